// TrainableGNNLayer_14113262535116
// MI455X (gfx1250) — compile-verified
//
#include <hip/hip_runtime.h>
#include <hip/hip_bf16.h>

typedef __attribute__((ext_vector_type(2))) float v2f;
typedef __attribute__((ext_vector_type(8))) float v8f;

#define DDIM   128          // feature dim (both in and out)
#define WSTR   132          // padded LDS row stride for W (conflict-free reads)

// ---------------------------------------------------------------------------
// Kernel 1: zero the aggregation buffer (d_ws), float4 grid-stride
// ---------------------------------------------------------------------------
__global__ void gnn_zero_kernel(float4* __restrict__ p, int n4) {
    int i = blockIdx.x * blockDim.x + threadIdx.x;
    int stride = gridDim.x * blockDim.x;
    float4 z = make_float4(0.f, 0.f, 0.f, 0.f);
    for (; i < n4; i += stride) p[i] = z;
}

// ---------------------------------------------------------------------------
// Kernel 2: edge gather + scatter-add.  One wave (32 lanes) per edge; each
// lane moves 4 floats: one float4 gather load + 4 native f32 global atomics.
// edge_index is [2, E]: row 0 = source, row 1 = dest.
// ---------------------------------------------------------------------------
__global__ __launch_bounds__(256)
void gnn_scatter_kernel(const float* __restrict__ nf,
                        const int* __restrict__ ei,
                        float* __restrict__ agg, int E) {
    int t    = blockIdx.x * blockDim.x + threadIdx.x;
    int e    = t >> 5;
    int lane = t & 31;
    if (e >= E) return;
    int src = ei[e];
    int dst = ei[E + e];
    float4 v = ((const float4*)nf)[src * (DDIM / 4) + lane];
    float* base = agg + (size_t)dst * DDIM + lane * 4;
    __hip_atomic_fetch_add(base + 0, v.x, __ATOMIC_RELAXED, __HIP_MEMORY_SCOPE_AGENT);
    __hip_atomic_fetch_add(base + 1, v.y, __ATOMIC_RELAXED, __HIP_MEMORY_SCOPE_AGENT);
    __hip_atomic_fetch_add(base + 2, v.z, __ATOMIC_RELAXED, __HIP_MEMORY_SCOPE_AGENT);
    __hip_atomic_fetch_add(base + 3, v.w, __ATOMIC_RELAXED, __HIP_MEMORY_SCOPE_AGENT);
}

// ---------------------------------------------------------------------------
// Kernel 3: out = agg @ W^T + b via V_WMMA_F32_16X16X4_F32, software-pipelined.
// 128 threads = 4 waves; each wave owns one 16-row tile and all 128 output
// columns (8 col-tiles -> 64 accumulator VGPRs).  W staged in LDS with a
// 132-float row stride so B-operand reads are bank-conflict-free.
// A-operand rows are per-lane float2 loads straight from L2-resident agg.
// Operand loads for k-step ks+1 are prefetched before the WMMAs of step ks.
// ---------------------------------------------------------------------------
__global__ __launch_bounds__(128)
void gnn_wmma_gemm_kernel(const float* __restrict__ agg,
                          const float* __restrict__ W,
                          const float* __restrict__ bias,
                          float* __restrict__ out, int nTiles) {
    extern __shared__ float sW[];                // DDIM rows x WSTR floats
    const int tid   = threadIdx.x;
    const int wave  = tid >> 5;
    const int lane  = tid & 31;
    const int mlo   = lane & 15;                 // M (or N) index within half-wave
    const int khalf = (lane >> 4) * 2;           // K sub-offset: 0 or 2

    // Cooperative staged copy of W (row-major, padded stride). 128 rows x 32 float4.
    for (int i = tid; i < DDIM * (DDIM / 4); i += 128) {
        int r = i >> 5, c = i & 31;
        ((float4*)(sW + r * WSTR))[c] = ((const float4*)W)[i];
    }
    __syncthreads();

    const int tile = blockIdx.x * 4 + wave;      // wave-uniform => EXEC all-ones
    if (tile >= nTiles) return;

    // Per-lane operand base pointers.
    const float2* aRow = (const float2*)(agg + (size_t)(tile * 16 + mlo) * DDIM);
    const int ahalf = lane >> 4;
    const float* sWB = sW + mlo * WSTR + khalf;  // + ot*16*WSTR + ks*4

    // Bias folded into accumulator init: every component of acc[ot] maps to
    // output column ot*16+mlo (rows differ), so init all 8 lanes' components.
    v8f acc[8];
    #pragma unroll
    for (int ot = 0; ot < 8; ++ot) {
        float bl = bias[ot * 16 + mlo];
        #pragma unroll
        for (int v = 0; v < 8; ++v) acc[ot][v] = bl;
    }

    // ---- software pipeline: prologue loads for ks = 0 ----
    v2f a_cur, b_cur[8], a_nxt, b_nxt[8];
    {
        float2 av = aRow[ahalf];
        a_cur.x = av.x; a_cur.y = av.y;
        #pragma unroll
        for (int ot = 0; ot < 8; ++ot) {
            float2 bv = *(const float2*)(sWB + ot * (16 * WSTR));
            b_cur[ot].x = bv.x; b_cur[ot].y = bv.y;
        }
    }

    #pragma unroll 2
    for (int ks = 0; ks < DDIM / 4 - 1; ++ks) {
        // prefetch operands for ks+1 (independent of this step's WMMAs)
        {
            float2 av = aRow[(ks + 1) * 2 + ahalf];
            a_nxt.x = av.x; a_nxt.y = av.y;
            const float* p = sWB + (ks + 1) * 4;
            #pragma unroll
            for (int ot = 0; ot < 8; ++ot) {
                float2 bv = *(const float2*)(p + ot * (16 * WSTR));
                b_nxt[ot].x = bv.x; b_nxt[ot].y = bv.y;
            }
        }
        // 8 back-to-back WMMAs on disjoint accumulators (no D->A/B hazard)
        #pragma unroll
        for (int ot = 0; ot < 8; ++ot) {
            acc[ot] = __builtin_amdgcn_wmma_f32_16x16x4_f32(
                false, a_cur, false, b_cur[ot], (short)0, acc[ot], false, false);
        }
        a_cur = a_nxt;
        #pragma unroll
        for (int ot = 0; ot < 8; ++ot) b_cur[ot] = b_nxt[ot];
    }
    // ---- epilogue: last k-step ----
    #pragma unroll
    for (int ot = 0; ot < 8; ++ot) {
        acc[ot] = __builtin_amdgcn_wmma_f32_16x16x4_f32(
            false, a_cur, false, b_cur[ot], (short)0, acc[ot], false, false);
    }

    // C/D layout: VGPR v holds row v (lanes 0-15) and row v+8 (lanes 16-31).
    const int rhalf8 = (lane >> 4) * 8;
    #pragma unroll
    for (int v = 0; v < 8; ++v) {
        const int row = tile * 16 + v + rhalf8;
        #pragma unroll
        for (int ot = 0; ot < 8; ++ot) {
            out[(size_t)row * DDIM + ot * 16 + mlo] = acc[ot][v];
        }
    }
}

// ---------------------------------------------------------------------------
// Host-side launch.  Inputs: node_features f32 [N,128], edge_index i32 [2,E],
// W f32 [128,128], b f32 [128].  Output f32 [N,128].  d_ws: f32 [N,128] accum.
// ---------------------------------------------------------------------------
extern "C" void kernel_launch(void* const* d_in, const int* in_sizes, int n_in,
                              void* d_out, int out_size, void* d_ws, size_t ws_size,
                              hipStream_t stream) {
    const float* node_features = (const float*)d_in[0];
    const int*   edge_index    = (const int*)d_in[1];
    const float* W             = (const float*)d_in[2];
    const float* b             = (const float*)d_in[3];
    float*       out           = (float*)d_out;
    float*       agg           = (float*)d_ws;

    const int nNodes = in_sizes[0] / DDIM;       // 50000
    const int E      = in_sizes[1] / 2;          // 800000
    const int nAgg   = nNodes * DDIM;            // 6.4M floats

    // 1) zero accumulator
    {
        int n4 = nAgg / 4;
        gnn_zero_kernel<<<2048, 256, 0, stream>>>((float4*)agg, n4);
    }
    // 2) gather + scatter-add (1 wave per edge)
    {
        long long threads = (long long)E * 32;
        int blocks = (int)((threads + 255) / 256);
        gnn_scatter_kernel<<<blocks, 256, 0, stream>>>(node_features, edge_index, agg, E);
    }
    // 3) WMMA GEMM + bias
    {
        int nTiles = nNodes / 16;                // 3125 (N divisible by 16)
        int blocks = (nTiles + 3) / 4;
        size_t smem = (size_t)DDIM * WSTR * sizeof(float);   // 67584 B
        gnn_wmma_gemm_kernel<<<blocks, 128, smem, stream>>>(agg, W, b, out, nTiles);
    }
}